// NUFFTLayerMultiChannel3D_Param_2662879723691
// MI455X (gfx1250) — compile-verified
//
#include <hip/hip_runtime.h>

// ---------------------------------------------------------------------------
// NUFFT layer for MI455X (gfx1250): separable Gaussian gridding + DFT-as-GEMM
// spectral filter + separable interpolation, all contractions on
// V_WMMA_F32_16X16X4_F32 (fp32 matrix pipe, wave32).
// ---------------------------------------------------------------------------

typedef float v2f __attribute__((ext_vector_type(2)));
typedef float v8f __attribute__((ext_vector_type(8)));

#define NG    32
#define N3    32768
#define BATCH 2
#define NP    512
#define NM    2
#define NC    2

constexpr float PI_F     = 3.14159265358979323846f;
constexpr float TAU_F    = 12.0f / (1024.0f * PI_F * PI_F);     // 12*(L/(2*pi*N))^2, L=2
constexpr float INV4TAU  = 1.0f / (4.0f * TAU_F);

__device__ __forceinline__ v8f wmma4(v2f a, v2f b, v8f c) {
  // D = A(16x4,f32) * B(4x16,f32) + C(16x16,f32)
  return __builtin_amdgcn_wmma_f32_16x16x4_f32(false, a, false, b, (short)0, c,
                                               false, false);
}

__device__ __forceinline__ float gterm(float d) {
  float a = d, b = d - 2.0f, c = d + 2.0f;
  return expf(-a * a * INV4TAU) + expf(-b * b * INV4TAU) + expf(-c * c * INV4TAU);
}

// line geometry for a DFT pass along one axis of a [32][32][32] array
__device__ __forceinline__ void axis_geom(int axis, int line, long& base, long& s) {
  if (axis == 0)      { s = 1024; base = line; }                               // contract d0
  else if (axis == 1) { s = 32;   base = (long)(line >> 5) * 1024 + (line & 31); } // d1
  else                { s = 1;    base = (long)line * 32; }                    // d2
}

// --------------------------- DFT matrices ---------------------------------
// mats layout (6 x 32 x 32 floats):
// [0]=Cf=cos  [1]=Sf=-sin  [2]=SfN=+sin  [3]=Ci=cos/32 [4]=Si=sin/32 [5]=SiN=-sin/32
__global__ void dftmat_k(float* mats) {
  int i = blockIdx.x * blockDim.x + threadIdx.x;
  if (i >= 1024) return;
  int nn = i >> 5, kk = i & 31;
  float ang = (2.0f * PI_F / 32.0f) * (float)(nn * kk);
  float c = cosf(ang), s = sinf(ang);
  mats[0 * 1024 + i] = c;
  mats[1 * 1024 + i] = -s;
  mats[2 * 1024 + i] = s;
  mats[3 * 1024 + i] = c * (1.0f / 32.0f);
  mats[4 * 1024 + i] = s * (1.0f / 32.0f);
  mats[5 * 1024 + i] = -s * (1.0f / 32.0f);
}

// --------------------------- geometry -------------------------------------
// One wave per (b,p). Lane i holds grid index i.
// GA [b*P+p][32]   : g0 = gterm(py - xv)      (axis0 factor, meshgrid 'ij')
// AX [b][32][P]    : fsum * g0               (gridding A matrix)
// G12[b*P+p][1024] : g1[d1]*g2[d2], g1 from px (axis1), g2 from pz (axis2)
__global__ void geom_k(const float* __restrict__ x, const float* __restrict__ f,
                       float* __restrict__ GA, float* __restrict__ AX,
                       float* __restrict__ G12) {
  int w = (blockIdx.x * blockDim.x + threadIdx.x) >> 5;
  int lane = threadIdx.x & 31;
  if (w >= BATCH * NP) return;
  int b = w >> 9, p = w & (NP - 1);
  float xv = -1.0f + (float)lane * 0.0625f;  // linspace(-1,1,33)[:-1]
  const float* xp = x + (long)w * 3;
  float px = xp[0], py = xp[1], pz = xp[2];
  float ga = gterm(py - xv);   // axis0
  float gb = gterm(px - xv);   // axis1
  float gc = gterm(pz - xv);   // axis2
  GA[(long)w * 32 + lane] = ga;
  float fsum = f[(long)(b * NC + 0) * NP + p] + f[(long)(b * NC + 1) * NP + p];
  AX[((long)b * 32 + lane) * NP + p] = fsum * ga;
  for (int i = 0; i < 32; ++i) {
    float gbi = __shfl(gb, i, 32);
    G12[(long)w * 1024 + i * 32 + lane] = gbi * gc;
  }
}

// --------------------------- generic fp32 WMMA GEMMs ----------------------
// D[M x N] = A[M x K] * B[K x N], all row-major, one 16x16 tile per wave.
__global__ void gemm_rr(const float* __restrict__ A, const float* __restrict__ Bm,
                        float* __restrict__ D, int Msize, int Nsize, int Ksize,
                        long strideA, long strideB, long strideD) {
  int lane = threadIdx.x & 31;
  int w = (blockIdx.x * blockDim.x + threadIdx.x) >> 5;
  int tiles_n = Nsize >> 4, tiles_m = Msize >> 4;
  if (w >= tiles_m * tiles_n) return;
  int mt = w / tiles_n, nt = w % tiles_n;
  const float* Ab = A + (long)blockIdx.z * strideA;
  const float* Bb = Bm + (long)blockIdx.z * strideB;
  float* Db = D + (long)blockIdx.z * strideD;
  int m = mt * 16 + (lane & 15);
  int n = nt * 16 + (lane & 15);
  int kh = (lane >> 4) << 1;
  v8f acc = {};
  for (int k0 = 0; k0 < Ksize; k0 += 4) {
    v2f a, b;
    a[0] = Ab[(long)m * Ksize + k0 + kh];
    a[1] = Ab[(long)m * Ksize + k0 + kh + 1];
    b[0] = Bb[(long)(k0 + kh) * Nsize + n];
    b[1] = Bb[(long)(k0 + kh + 1) * Nsize + n];
    acc = wmma4(a, b, acc);
  }
  int row0 = mt * 16 + 8 * (lane >> 4);
  for (int j = 0; j < 8; ++j)
    Db[(long)(row0 + j) * Nsize + n] = acc[j];
}

// D[M x N] = A[M x K] * B^T, B stored row-major [N x K]. B batch = z / bdiv.
__global__ void gemm_ABt(const float* __restrict__ A, const float* __restrict__ Bm,
                         float* __restrict__ D, int Msize, int Nsize, int Ksize,
                         long strideA, long strideB, long strideD, int bdiv) {
  int lane = threadIdx.x & 31;
  int w = (blockIdx.x * blockDim.x + threadIdx.x) >> 5;
  int tiles_n = Nsize >> 4, tiles_m = Msize >> 4;
  if (w >= tiles_m * tiles_n) return;
  int mt = w / tiles_n, nt = w % tiles_n;
  const float* Ab = A + (long)blockIdx.z * strideA;
  const float* Bb = Bm + (long)(blockIdx.z / bdiv) * strideB;
  float* Db = D + (long)blockIdx.z * strideD;
  int m = mt * 16 + (lane & 15);
  int n = nt * 16 + (lane & 15);
  int kh = (lane >> 4) << 1;
  v8f acc = {};
  for (int k0 = 0; k0 < Ksize; k0 += 4) {
    v2f a, b;
    a[0] = Ab[(long)m * Ksize + k0 + kh];
    a[1] = Ab[(long)m * Ksize + k0 + kh + 1];
    b[0] = Bb[(long)n * Ksize + k0 + kh];
    b[1] = Bb[(long)n * Ksize + k0 + kh + 1];
    acc = wmma4(a, b, acc);
  }
  int row0 = mt * 16 + 8 * (lane >> 4);
  for (int j = 0; j < 8; ++j)
    Db[(long)(row0 + j) * Nsize + n] = acc[j];
}

// --------------------------- DFT passes (WMMA) ----------------------------
// 1024 lines x 32 outputs per array; 128 wave-tiles; batch = blockIdx.z.
__global__ void dft_r2c(const float* __restrict__ in, float* __restrict__ ore,
                        float* __restrict__ oim, const float* __restrict__ m3,
                        int axis) {
  const float* Cm = m3;
  const float* Sm = m3 + 1024;
  int lane = threadIdx.x & 31;
  int w = (blockIdx.x * blockDim.x + threadIdx.x) >> 5;
  if (w >= 128) return;
  int mt = w >> 1, nt = w & 1;
  long boff = (long)blockIdx.z * N3;
  int lm = mt * 16 + (lane & 15);
  int kh = (lane >> 4) << 1;
  int n = nt * 16 + (lane & 15);
  long base, s; axis_geom(axis, lm, base, s);
  const float* inb = in + boff;
  v8f ar = {}, ai = {};
  for (int k0 = 0; k0 < 32; k0 += 4) {
    v2f a, bc, bs;
    a[0] = inb[base + (long)(k0 + kh) * s];
    a[1] = inb[base + (long)(k0 + kh + 1) * s];
    bc[0] = Cm[(k0 + kh) * 32 + n]; bc[1] = Cm[(k0 + kh + 1) * 32 + n];
    bs[0] = Sm[(k0 + kh) * 32 + n]; bs[1] = Sm[(k0 + kh + 1) * 32 + n];
    ar = wmma4(a, bc, ar);
    ai = wmma4(a, bs, ai);
  }
  int half = lane >> 4;
  for (int j = 0; j < 8; ++j) {
    int line = mt * 16 + j + 8 * half;
    long ob, os; axis_geom(axis, line, ob, os);
    ore[boff + ob + (long)n * os] = ar[j];
    oim[boff + ob + (long)n * os] = ai[j];
  }
}

// complex->complex axis pass: out_re = re*C + im*SN ; out_im = re*S + im*C
__global__ void dft_c2c(const float* __restrict__ ire, const float* __restrict__ iim,
                        float* __restrict__ ore, float* __restrict__ oim,
                        const float* __restrict__ m3, int axis) {
  const float* Cm = m3;
  const float* Sm = m3 + 1024;
  const float* SmN = m3 + 2048;
  int lane = threadIdx.x & 31;
  int w = (blockIdx.x * blockDim.x + threadIdx.x) >> 5;
  if (w >= 128) return;
  int mt = w >> 1, nt = w & 1;
  long boff = (long)blockIdx.z * N3;
  int lm = mt * 16 + (lane & 15);
  int kh = (lane >> 4) << 1;
  int n = nt * 16 + (lane & 15);
  long base, s; axis_geom(axis, lm, base, s);
  const float* re = ire + boff;
  const float* im = iim + boff;
  v8f ar = {}, ai = {};
  for (int k0 = 0; k0 < 32; k0 += 4) {
    v2f a0, a1, bc, bs, bn;
    long o0 = base + (long)(k0 + kh) * s, o1 = base + (long)(k0 + kh + 1) * s;
    a0[0] = re[o0]; a0[1] = re[o1];
    a1[0] = im[o0]; a1[1] = im[o1];
    bc[0] = Cm[(k0 + kh) * 32 + n];  bc[1] = Cm[(k0 + kh + 1) * 32 + n];
    bs[0] = Sm[(k0 + kh) * 32 + n];  bs[1] = Sm[(k0 + kh + 1) * 32 + n];
    bn[0] = SmN[(k0 + kh) * 32 + n]; bn[1] = SmN[(k0 + kh + 1) * 32 + n];
    ar = wmma4(a0, bc, ar); ar = wmma4(a1, bn, ar);
    ai = wmma4(a0, bs, ai); ai = wmma4(a1, bc, ai);
  }
  int half = lane >> 4;
  for (int j = 0; j < 8; ++j) {
    int line = mt * 16 + j + 8 * half;
    long ob, os; axis_geom(axis, line, ob, os);
    ore[boff + ob + (long)n * os] = ar[j];
    oim[boff + ob + (long)n * os] = ai[j];
  }
}

// complex->real final inverse pass: out = re*Ci + im*SiN
__global__ void dft_c2r(const float* __restrict__ ire, const float* __restrict__ iim,
                        float* __restrict__ ore, const float* __restrict__ m3,
                        int axis) {
  const float* Cm = m3;
  const float* SmN = m3 + 2048;
  int lane = threadIdx.x & 31;
  int w = (blockIdx.x * blockDim.x + threadIdx.x) >> 5;
  if (w >= 128) return;
  int mt = w >> 1, nt = w & 1;
  long boff = (long)blockIdx.z * N3;
  int lm = mt * 16 + (lane & 15);
  int kh = (lane >> 4) << 1;
  int n = nt * 16 + (lane & 15);
  long base, s; axis_geom(axis, lm, base, s);
  const float* re = ire + boff;
  const float* im = iim + boff;
  v8f ar = {};
  for (int k0 = 0; k0 < 32; k0 += 4) {
    v2f a0, a1, bc, bn;
    long o0 = base + (long)(k0 + kh) * s, o1 = base + (long)(k0 + kh + 1) * s;
    a0[0] = re[o0]; a0[1] = re[o1];
    a1[0] = im[o0]; a1[1] = im[o1];
    bc[0] = Cm[(k0 + kh) * 32 + n];  bc[1] = Cm[(k0 + kh + 1) * 32 + n];
    bn[0] = SmN[(k0 + kh) * 32 + n]; bn[1] = SmN[(k0 + kh + 1) * 32 + n];
    ar = wmma4(a0, bc, ar); ar = wmma4(a1, bn, ar);
  }
  int half = lane >> 4;
  for (int j = 0; j < 8; ++j) {
    int line = mt * 16 + j + 8 * half;
    long ob, os; axis_geom(axis, line, ob, os);
    ore[boff + ob + (long)n * os] = ar[j];
  }
}

// --------------------------- k-space filter --------------------------------
// H[b,m,k] = F[b,k] * deconv(k) * mult[m](k), fftshift folded into (d+16)&31.
__global__ void filter_k(const float* __restrict__ Fre, const float* __restrict__ Fim,
                         const float* __restrict__ p_sh, const float* __restrict__ p_am,
                         const float* __restrict__ p_be, const float* __restrict__ p_hy,
                         float* __restrict__ Hre, float* __restrict__ Him) {
  int i = blockIdx.x * blockDim.x + threadIdx.x;
  if (i >= N3) return;
  int b = blockIdx.z;
  int d0 = i >> 10, d1 = (i >> 5) & 31, d2 = i & 31;
  int s0 = (d0 + 16) & 31, s1 = (d1 + 16) & 31, s2 = (d2 + 16) & 31;
  float k0v = PI_F * (-16.0f + (32.0f / 31.0f) * (float)s0);
  float k1v = PI_F * (-16.0f + (32.0f / 31.0f) * (float)s1);
  float k2v = PI_F * (-16.0f + (32.0f / 31.0f) * (float)s2);
  float k2 = k0v * k0v + k1v * k1v + k2v * k2v;
  float kmag = sqrtf(k2);
  float pot = PI_F / TAU_F;
  float dec = pot * sqrtf(pot) * expf(k2 * TAU_F);
  float fr = Fre[(long)b * N3 + i] * dec;
  float fi = Fim[(long)b * N3 + i] * dec;
  for (int m = 0; m < NM; ++m) {
    float mult = 0.0f;
    for (int c = 0; c < NC; ++c) {
      float sh = p_sh[m * NC + c], am = p_am[m * NC + c];
      float be = p_be[m * NC + c], hy = p_hy[m * NC + c];
      float a2 = am * am;
      float t1 = kmag + sh, t2 = kmag - sh;
      float part = am / (t1 * t1 + a2) + am / (t2 * t2 + a2);
      float filt = k2 / (k2 + 400.0f * hy);
      mult += be * part * filt;
    }
    Hre[(long)(b * NM + m) * N3 + i] = fr * mult;
    Him[(long)(b * NM + m) * N3 + i] = fi * mult;
  }
}

// --------------------------- final contraction -----------------------------
// energy[b,m,p] = sum_d0 GA[b,p,d0] * T[b,m,d0,p]
__global__ void energy_k(const float* __restrict__ GA, const float* __restrict__ T,
                         float* __restrict__ out) {
  int idx = blockIdx.x * blockDim.x + threadIdx.x;
  if (idx >= BATCH * NM * NP) return;
  int p = idx & (NP - 1);
  int bm = idx >> 9;
  int b = bm >> 1;
  const float* ga = GA + ((long)(b * NP + p)) * 32;
  const float* t = T + (long)bm * 32 * NP + p;
  float s = 0.0f;
  for (int d0 = 0; d0 < 32; ++d0) s += ga[d0] * t[(long)d0 * NP];
  out[idx] = s;
}

// --------------------------- launch ---------------------------------------
extern "C" void kernel_launch(void* const* d_in, const int* in_sizes, int n_in,
                              void* d_out, int out_size, void* d_ws, size_t ws_size,
                              hipStream_t stream) {
  (void)in_sizes; (void)n_in; (void)out_size;
  const float* x     = (const float*)d_in[0];
  const float* f     = (const float*)d_in[1];
  const float* shift = (const float*)d_in[2];
  const float* amp   = (const float*)d_in[3];
  const float* beta  = (const float*)d_in[4];
  const float* hyp   = (const float*)d_in[5];
  float* out = (float*)d_out;
  float* ws  = (float*)d_ws;

  float* GA   = ws + 0;        //  32768  [b*P+p][32]
  float* AX   = ws + 32768;    //  32768  [b][32][P]
  float* G12  = ws + 65536;    // 1048576 [b*P+p][1024]
  float* GRID = ws + 1114112;  //  65536  [b][32][1024]
  float* FAre = ws + 1179648;  //  65536  (ping)
  float* FAim = ws + 1245184;
  float* FBre = ws + 1310720;  //  (pong)
  float* FBim = ws + 1376256;
  float* HAre = ws + 1441792;  // 131072  [b*M+m][32768]
  float* HAim = ws + 1572864;
  float* HBre = ws + 1703936;
  float* HBim = ws + 1835008;
  float* INV  = ws + 1966080;  // 131072
  float* T    = ws + 2097152;  //  65536  [b*M+m][32][P]
  float* MATS = ws + 2162688;  //   6144
  if (ws_size < (size_t)2168832 * sizeof(float)) return;

  dftmat_k<<<4, 256, 0, stream>>>(MATS);
  geom_k<<<128, 256, 0, stream>>>(x, f, GA, AX, G12);

  // gridding: grid[b] (32x1024) = AX[b] (32x512) * G12[b] (512x1024)
  gemm_rr<<<dim3(32, 1, BATCH), 128, 0, stream>>>(
      AX, G12, GRID, 32, 1024, 512, (long)32 * 512, (long)512 * 1024, (long)N3);

  // forward 3D DFT (real input): axes d2, d1, d0
  dft_r2c<<<dim3(32, 1, BATCH), 128, 0, stream>>>(GRID, FAre, FAim, MATS, 2);
  dft_c2c<<<dim3(32, 1, BATCH), 128, 0, stream>>>(FAre, FAim, FBre, FBim, MATS, 1);
  dft_c2c<<<dim3(32, 1, BATCH), 128, 0, stream>>>(FBre, FBim, FAre, FAim, MATS, 0);

  // k-space filter (deconv * per-m multiplier, shift folded into indexing)
  filter_k<<<dim3(128, 1, BATCH), 256, 0, stream>>>(FAre, FAim, shift, amp, beta,
                                                    hyp, HAre, HAim);

  // inverse 3D DFT on B*M fields; final pass keeps real part only
  dft_c2c<<<dim3(32, 1, BATCH * NM), 128, 0, stream>>>(HAre, HAim, HBre, HBim,
                                                       MATS + 3072, 2);
  dft_c2c<<<dim3(32, 1, BATCH * NM), 128, 0, stream>>>(HBre, HBim, HAre, HAim,
                                                       MATS + 3072, 1);
  dft_c2r<<<dim3(32, 1, BATCH * NM), 128, 0, stream>>>(HAre, HAim, INV,
                                                       MATS + 3072, 0);

  // interpolation: T[b,m] (32x512) = INV[b,m] (32x1024) * G12[b]^T
  gemm_ABt<<<dim3(16, 1, BATCH * NM), 128, 0, stream>>>(
      INV, G12, T, 32, 512, 1024, (long)N3, (long)512 * 1024, (long)32 * 512, NM);

  energy_k<<<8, 256, 0, stream>>>(GA, T, out);
}